// STN_Polar_71116068488022
// MI455X (gfx1250) — compile-verified
//
#include <hip/hip_runtime.h>
#include <stdint.h>

#define N_  16
#define C_  32
#define H_  512
#define W_  512
#define S_  256

// Unaligned-capable float2: lets clang emit one global_load_b64 at 4-byte
// alignment (legal in DWORD alignment mode) for the (v0,v1) x-pair gathers.
typedef float f2_t __attribute__((ext_vector_type(2)));
typedef f2_t uf2_t __attribute__((aligned(4)));

// One block per (n, i) row of the log-polar grid; threadIdx.x = j (rho index).
// 256 threads = 8 wave32s. theta is uniform per block (scalarized).
__global__ __launch_bounds__(256) void stn_polar_kernel(
    const float* __restrict__ x,      // [16,32,512,512]
    const float* __restrict__ polar,  // [16,2]
    float* __restrict__ out,          // [16,32,256,256]
    float* __restrict__ grid)         // [16,256,256,2]
{
    __shared__ float2 sgrid[256];

    const int j = threadIdx.x;
    const int n = blockIdx.x >> 8;
    const int i = blockIdx.x & 255;

    // uniform per block -> scalar loads
    const float px = polar[n * 2 + 0];
    const float py = polar[n * 2 + 1];

    // rho = exp(ln(128)/256 * j) - 1 ; theta = i * 2pi/256
    const float mag = 0.0189532432184360f;   // ln(128)/256
    const float dth = 0.0245436926061703f;   // 2*pi/256
    float rho   = __expf(mag * (float)j) - 1.0f;
    float theta = dth * (float)i;
    float sn, cs;
    __sincosf(theta, &sn, &cs);

    const float gx = (rho * cs + px) * (1.0f / 256.0f);
    const float gy = (rho * sn + py) * (1.0f / 256.0f);

    // bilinear source coords, padding_mode='border', align_corners=False
    float ixf = fminf(fmaxf(fmaf(gx, 256.0f, 255.5f), 0.0f), 511.0f);
    float iyf = fminf(fmaxf(fmaf(gy, 256.0f, 255.5f), 0.0f), 511.0f);
    float x0f = floorf(ixf);
    float y0f = floorf(iyf);
    const float wx = ixf - x0f;
    const float wy = iyf - y0f;
    const int x0 = (int)x0f;
    const int y0 = (int)y0f;
    const int y1 = min(y0 + 1, H_ - 1);
    // x1 is implicitly x0+1 via the paired b64 load; when the x-clamp would
    // engage (x0==511), wx==0 so element [1] is a don't-care.

    // ---- grid output: LDS stage -> async store-from-LDS (ASYNCcnt path),
    // non-temporal so the 8 MB write-once stream doesn't occupy L2 ----
    sgrid[j] = make_float2(gx, gy);
    {
        float* gp = grid + ((((size_t)n * S_ + i) * S_) + j) * 2;
        unsigned lds_off = (unsigned)(uintptr_t)(&sgrid[j]); // low 32b of flat = LDS byte offset
        asm volatile("s_wait_dscnt 0" ::: "memory");         // DS write -> async read RAW
        asm volatile("global_store_async_from_lds_b64 %0, %1, off th:TH_STORE_NT"
                     :: "v"((uint64_t)(uintptr_t)gp), "v"(lds_off)
                     : "memory");
        // s_endpgm performs an implicit wait-idle, covering ASYNCcnt.
    }

    // ---- main gather loop over channels: coords amortized over all 32 ----
    const float* __restrict__ base = x + (size_t)n * (C_ * H_ * W_);
    float* __restrict__ op = out + (size_t)n * (C_ * S_ * S_) + (size_t)i * S_ + j;

    const size_t oTop = (size_t)(y0 * W_ + x0);   // (v00,v01) pair
    const size_t oBot = (size_t)(y1 * W_ + x0);   // (v10,v11) pair

    // Peeled: channels 0..23 prefetch 8 ahead (1 MB channel stride ->
    // guaranteed L0 miss; global_prefetch_b8 turns it into a WGP$ hit).
#pragma unroll 8
    for (int c = 0; c < C_ - 8; ++c) {
        const float* __restrict__ p  = base + (size_t)c * (H_ * W_);
        const float* __restrict__ pf = base + (size_t)(c + 8) * (H_ * W_);
        __builtin_prefetch(pf + oTop, 0, 1);
        __builtin_prefetch(pf + oBot, 0, 1);

        const f2_t vt = *(const uf2_t*)(p + oTop);   // one b64 gather: v00,v01
        const f2_t vb = *(const uf2_t*)(p + oBot);   // one b64 gather: v10,v11

        const float top = fmaf(wx, vt.y - vt.x, vt.x);
        const float bot = fmaf(wx, vb.y - vb.x, vb.x);
        const float r   = fmaf(wy, bot - top, top);

        // Non-temporal store: 142 MB write-once stream must not evict the
        // ~100 MB gather-hot input set from the 192 MB L2.
        __builtin_nontemporal_store(r, op + (size_t)c * (S_ * S_));
    }
    // Tail: channels 24..31, already prefetched.
#pragma unroll 8
    for (int c = C_ - 8; c < C_; ++c) {
        const float* __restrict__ p = base + (size_t)c * (H_ * W_);

        const f2_t vt = *(const uf2_t*)(p + oTop);
        const f2_t vb = *(const uf2_t*)(p + oBot);

        const float top = fmaf(wx, vt.y - vt.x, vt.x);
        const float bot = fmaf(wx, vb.y - vb.x, vb.x);
        const float r   = fmaf(wy, bot - top, top);

        __builtin_nontemporal_store(r, op + (size_t)c * (S_ * S_));
    }
}

extern "C" void kernel_launch(void* const* d_in, const int* in_sizes, int n_in,
                              void* d_out, int out_size, void* d_ws, size_t ws_size,
                              hipStream_t stream) {
    (void)in_sizes; (void)n_in; (void)out_size; (void)d_ws; (void)ws_size;

    const float* x     = (const float*)d_in[0];   // [16,32,512,512] fp32
    const float* polar = (const float*)d_in[1];   // [16,2] fp32

    float* out  = (float*)d_out;                               // 16*32*256*256
    float* grid = out + (size_t)N_ * C_ * S_ * S_;             // then 16*256*256*2

    dim3 blocks(N_ * S_);   // 4096: one per (n, i)
    dim3 threads(S_);       // 256: j across the row -> coalesced stores
    hipLaunchKernelGGL(stn_polar_kernel, blocks, threads, 0, stream,
                       x, polar, out, grid);
}